// DMoNPooling_25383256720124
// MI455X (gfx1250) — compile-verified
//
#include <hip/hip_runtime.h>
#include <math.h>

typedef __attribute__((ext_vector_type(2))) float v2f;
typedef __attribute__((ext_vector_type(8))) float v8f;

#define B_ 8
#define N_ 2048
#define F_ 256
#define C_ 64

// D = A(16x4, f32) * B(4x16, f32) + C(16x16, f32)   [V_WMMA_F32_16X16X4_F32]
__device__ __forceinline__ v8f wmma_f32(v2f a, v2f b, v8f c) {
  return __builtin_amdgcn_wmma_f32_16x16x4_f32(false, a, false, b, (short)0, c,
                                               false, false);
}

__device__ __forceinline__ v8f vzero8() {
  v8f z;
#pragma unroll
  for (int i = 0; i < 8; ++i) z[i] = 0.0f;
  return z;
}

// -------------------- Kernel 1: s = softmax(x@W + b) * mask --------------------
// grid = (B*N)/32 = 512 blocks, 256 threads (8 waves). 32 rows x 64 cols per block.
__global__ void softmax_kernel(const float* __restrict__ x, const float* __restrict__ W,
                               const float* __restrict__ bias,
                               const unsigned char* __restrict__ mask,
                               float* __restrict__ s_out) {
  __shared__ float lg[32 * 64];
  __shared__ float bl[64];
  const int tid = threadIdx.x;
  const int wave = tid >> 5, lane = tid & 31;
  const int m = lane & 15, h = lane >> 4;
  const long row0 = (long)blockIdx.x * 32;
  if (tid < 64) bl[tid] = bias[tid];
  const int rt = wave >> 2;  // 0..1 (16-row tile)
  const int ct = wave & 3;   // 0..3 (16-col tile)
  v8f acc = vzero8();
  const float* __restrict__ xrow = x + (row0 + rt * 16 + m) * (long)F_;
  for (int k0 = 0; k0 < F_; k0 += 4) {
    const int ka = k0 + 2 * h;
    const float2 av = *(const float2*)(xrow + ka);
    v2f a; a.x = av.x; a.y = av.y;
    v2f bv;
    bv.x = W[ka * C_ + ct * 16 + m];
    bv.y = W[(ka + 1) * C_ + ct * 16 + m];
    acc = wmma_f32(a, bv, acc);
  }
  __syncthreads();  // bl ready
#pragma unroll
  for (int r = 0; r < 8; ++r)
    lg[(rt * 16 + r + h * 8) * C_ + ct * 16 + m] = acc[r] + bl[ct * 16 + m];
  __syncthreads();
  if (tid < 32) {  // one wave: one row each
    const int r = tid;
    const long grow = row0 + r;
    float mx = -3.4e38f;
    for (int c = 0; c < C_; ++c) mx = fmaxf(mx, lg[r * C_ + c]);
    float sum = 0.f;
    for (int c = 0; c < C_; ++c) sum += expf(lg[r * C_ + c] - mx);
    const float inv = 1.0f / sum;
    const float mv = mask[grow] ? 1.0f : 0.0f;
    for (int c = 0; c < C_; ++c)
      s_out[grow * C_ + c] = expf(lg[r * C_ + c] - mx) * inv * mv;
  }
}

// ---------- Kernel 2: t = adj @ s, fused degrees = rowsum(adj)*mask ----------
// grid = B*16 = 128 blocks, 256 threads (8 waves). Each wave: 16 rows x 64 cols, K=2048.
// s is staged in LDS per 128-row K-chunk (stride 72 floats: 2*72 mod 64 == 16, so the
// two K-halves of a wave hit disjoint LDS bank groups -> conflict-free ds_load).
#define KC_ 128
#define SP_ 72
__global__ void adj_s_kernel(const float* __restrict__ adj, const float* __restrict__ s,
                             const unsigned char* __restrict__ mask,
                             float* __restrict__ t, float* __restrict__ degrees) {
  __shared__ float sl[KC_ * SP_];  // 36 KB
  const int b = blockIdx.x >> 4;
  const int rb = blockIdx.x & 15;
  const int tid = threadIdx.x;
  const int wave = tid >> 5, lane = tid & 31;
  const int m = lane & 15, h = lane >> 4;
  const int rowbase = rb * 128 + wave * 16;
  const float* __restrict__ S = s + (size_t)b * N_ * C_;
  const float* __restrict__ arow =
      adj + ((size_t)b * N_ + rowbase + m) * N_;
  v8f acc0 = vzero8(), acc1 = vzero8(), acc2 = vzero8(), acc3 = vzero8();
  float rs = 0.f;
  for (int kc = 0; kc < N_; kc += KC_) {
    // cooperative, coalesced stage of s[kc..kc+KC_) into padded LDS
    for (int i = tid; i < KC_ * 16; i += 256) {
      const int r = i >> 4;
      const int q = (i & 15) << 2;
      const float4 v4 = *(const float4*)(S + (size_t)(kc + r) * C_ + q);
      *(float4*)(sl + r * SP_ + q) = v4;
    }
    __syncthreads();
    if (kc + KC_ < N_)  // prefetch next chunk of this lane's adj row
      __builtin_prefetch(arow + kc + KC_, 0, 0);
    for (int k0 = 0; k0 < KC_; k0 += 4) {
      const int ka = k0 + 2 * h;  // K index within chunk for this half-wave
      const float2 av = *(const float2*)(arow + kc + ka);
      v2f a; a.x = av.x; a.y = av.y;
      rs += av.x + av.y;
      const float* __restrict__ s0 = sl + ka * SP_ + m;
      const float* __restrict__ s1 = s0 + SP_;
      v2f b0; b0.x = s0[0];  b0.y = s1[0];
      acc0 = wmma_f32(a, b0, acc0);
      v2f b1; b1.x = s0[16]; b1.y = s1[16];
      acc1 = wmma_f32(a, b1, acc1);
      v2f b2; b2.x = s0[32]; b2.y = s1[32];
      acc2 = wmma_f32(a, b2, acc2);
      v2f b3; b3.x = s0[48]; b3.y = s1[48];
      acc3 = wmma_f32(a, b3, acc3);
    }
    __syncthreads();
  }
  // combine K-halves of the row sum: lane L (h=0) + lane L+16 (h=1)
  rs += __shfl_xor(rs, 16);
  if (h == 0) {
    const size_t gi = (size_t)b * N_ + rowbase + m;
    degrees[gi] = mask[gi] ? rs : 0.0f;
  }
  float* __restrict__ T = t + ((size_t)b * N_ + rowbase) * C_;
#pragma unroll
  for (int r = 0; r < 8; ++r) {
    const int row = r + h * 8;
    T[row * C_ + m]      = acc0[r];
    T[row * C_ + 16 + m] = acc1[r];
    T[row * C_ + 32 + m] = acc2[r];
    T[row * C_ + 48 + m] = acc3[r];
  }
}

// -------------------- Kernel 3: out = selu(s^T @ x) --------------------
// grid = B*4 blocks (batch, f-quarter), 256 threads. Wave w: c-tile = w>>1,
// f-tiles = (w&1)*2 + {0,1} within the 64-col quarter.
__global__ void stx_kernel(const float* __restrict__ s, const float* __restrict__ x,
                           float* __restrict__ out) {
  const int b = blockIdx.x >> 2;
  const int fq = blockIdx.x & 3;
  const int fbase0 = fq * 64;
  const int tid = threadIdx.x;
  const int wave = tid >> 5, lane = tid & 31;
  const int m = lane & 15, h = lane >> 4;
  const int ct = wave >> 1;
  const int ftp = (wave & 1) * 2;
  const float* __restrict__ S = s + (size_t)b * N_ * C_;
  const float* __restrict__ X = x + (size_t)b * N_ * F_;
  v8f acc[2];
#pragma unroll
  for (int i = 0; i < 2; ++i) acc[i] = vzero8();
  for (int k0 = 0; k0 < N_; k0 += 4) {
    const int ka = k0 + 2 * h;
    v2f a;  // A = s^T : A[c][k] = s[k][c]
    a.x = S[ka * C_ + ct * 16 + m];
    a.y = S[(ka + 1) * C_ + ct * 16 + m];
    const float* __restrict__ x0 = X + (size_t)ka * F_ + fbase0 + m;
    const float* __restrict__ x1 = x0 + F_;
#pragma unroll
    for (int j = 0; j < 2; ++j) {
      const int ft = ftp + j;
      v2f bv; bv.x = x0[ft * 16]; bv.y = x1[ft * 16];
      acc[j] = wmma_f32(a, bv, acc[j]);
    }
  }
  const float kScale = 1.0507009873554805f;
  const float kAlpha = 1.6732632423543772f;
  float* __restrict__ O = out + (size_t)b * C_ * F_;
#pragma unroll
  for (int j = 0; j < 2; ++j) {
#pragma unroll
    for (int r = 0; r < 8; ++r) {
      const int c = ct * 16 + r + h * 8;
      const int f = fbase0 + (ftp + j) * 16 + m;
      const float z = acc[j][r];
      O[c * F_ + f] = kScale * (z > 0.f ? z : kAlpha * expm1f(z));
    }
  }
}

// ------------- Kernel 4: oa_raw = s^T @ t ; ssm = s^T @ s -------------
// grid = B*2 blocks (batch, which), 256 threads. Wave w: c-tile = w>>1,
// n-tiles = (w&1)*2 + {0,1}.
__global__ void stt_ss_kernel(const float* __restrict__ s, const float* __restrict__ t,
                              float* __restrict__ oa_raw, float* __restrict__ ssm) {
  const int b = blockIdx.x >> 1;
  const int which = blockIdx.x & 1;  // 0: oa = s^T@t, 1: ss = s^T@s
  const int tid = threadIdx.x;
  const int wave = tid >> 5, lane = tid & 31;
  const int m = lane & 15, h = lane >> 4;
  const int ct = wave >> 1;
  const int ntp = (wave & 1) * 2;
  const float* __restrict__ S = s + (size_t)b * N_ * C_;
  const float* __restrict__ Bm = which ? S : (t + (size_t)b * N_ * C_);
  v8f acc[2];
#pragma unroll
  for (int i = 0; i < 2; ++i) acc[i] = vzero8();
  for (int k0 = 0; k0 < N_; k0 += 4) {
    const int ka = k0 + 2 * h;
    v2f a;
    a.x = S[ka * C_ + ct * 16 + m];
    a.y = S[(ka + 1) * C_ + ct * 16 + m];
    const float* __restrict__ b0 = Bm + ka * C_ + m;
    const float* __restrict__ b1 = Bm + (ka + 1) * C_ + m;
#pragma unroll
    for (int j = 0; j < 2; ++j) {
      const int nt = ntp + j;
      v2f bv; bv.x = b0[nt * 16]; bv.y = b1[nt * 16];
      acc[j] = wmma_f32(a, bv, acc[j]);
    }
  }
  float* __restrict__ D = (which ? ssm : oa_raw) + (size_t)b * C_ * C_;
#pragma unroll
  for (int j = 0; j < 2; ++j)
#pragma unroll
    for (int r = 0; r < 8; ++r)
      D[(ct * 16 + r + h * 8) * C_ + (ntp + j) * 16 + m] = acc[j][r];
}

// ------ Kernel 5: per-batch reductions: m, Nvalid, ca = s^T deg, cs = colsum(s) ------
__global__ void reduce_kernel(const float* __restrict__ s, const float* __restrict__ degrees,
                              const unsigned char* __restrict__ mask,
                              float* __restrict__ ca, float* __restrict__ cs,
                              float* __restrict__ msum, float* __restrict__ nvalid) {
  __shared__ float red[256];
  __shared__ float red2[256];
  const int b = blockIdx.x, tid = threadIdx.x;
  float dsum = 0.f, nv = 0.f;
  for (int n = tid; n < N_; n += 256) {
    dsum += degrees[b * N_ + n];
    nv += mask[b * N_ + n] ? 1.0f : 0.0f;
  }
  red[tid] = dsum; red2[tid] = nv;
  __syncthreads();
  for (int off = 128; off > 0; off >>= 1) {
    if (tid < off) { red[tid] += red[tid + off]; red2[tid] += red2[tid + off]; }
    __syncthreads();
  }
  if (tid == 0) { msum[b] = red[0] * 0.5f; nvalid[b] = red2[0]; }
  __syncthreads();
  const int c = tid & 63, g = tid >> 6;
  float cav = 0.f, csv = 0.f;
  for (int n = g; n < N_; n += 4) {
    const float sv = s[((size_t)b * N_ + n) * C_ + c];
    const float dg = degrees[b * N_ + n];
    csv += sv;
    cav += sv * dg;
  }
  red[tid] = cav; red2[tid] = csv;
  __syncthreads();
  if (tid < 64) {
    ca[b * C_ + tid] = red[tid] + red[tid + 64] + red[tid + 128] + red[tid + 192];
    cs[b * C_ + tid] = red2[tid] + red2[tid + 64] + red2[tid + 128] + red2[tid + 192];
  }
}

// ------ Kernel 6: losses + degree-normalized out_adj. 1 block, wave b -> batch b ------
__global__ void finalize_kernel(const float* __restrict__ oa_raw, const float* __restrict__ ssm,
                                const float* __restrict__ ca, const float* __restrict__ cs,
                                const float* __restrict__ msum, const float* __restrict__ nvalid,
                                float* __restrict__ oadj_out, float* __restrict__ scal_out) {
  __shared__ float dbuf[8][64];
  __shared__ float spec[8], orth[8], csn_s[8], nvi[8];
  const int tid = threadIdx.x;
  const int b = tid >> 5, lane = tid & 31;
  const float* __restrict__ OA = oa_raw + b * C_ * C_;
  const float* __restrict__ SS = ssm + b * C_ * C_;
  float tr = 0.f, ss2 = 0.f, trss = 0.f;
  for (int i = lane; i < C_ * C_; i += 32) {
    const int r = i >> 6, cc = i & 63;
    const float oav = OA[i];
    const float sv = SS[i];
    if (r == cc) { tr += oav; trss += sv; }
    ss2 += sv * sv;
  }
  for (int o = 16; o > 0; o >>= 1) {
    tr += __shfl_xor(tr, o);
    ss2 += __shfl_xor(ss2, o);
    trss += __shfl_xor(trss, o);
  }
  float ca2 = 0.f, cs2 = 0.f;
  for (int i = lane; i < C_; i += 32) {
    const float cv = ca[b * C_ + i];  ca2 += cv * cv;
    const float c2 = cs[b * C_ + i];  cs2 += c2 * c2;
  }
  for (int o = 16; o > 0; o >>= 1) {
    ca2 += __shfl_xor(ca2, o);
    cs2 += __shfl_xor(cs2, o);
  }
  // d[r] = sqrt(rowsum of diag-zeroed oa) + EPS
  for (int rr = lane; rr < C_; rr += 32) {
    float rsum = 0.f;
    for (int j = 0; j < C_; ++j)
      if (j != rr) rsum += OA[rr * C_ + j];
    dbuf[b][rr] = sqrtf(rsum) + 1e-15f;
  }
  __syncthreads();
  for (int i = lane; i < C_ * C_; i += 32) {
    const int r = i >> 6, cc = i & 63;
    const float v = (r == cc) ? 0.f : OA[i] / (dbuf[b][cc] * dbuf[b][r]);
    oadj_out[b * C_ * C_ + i] = v;
  }
  if (lane == 0) {
    const float mb = msum[b];
    spec[b] = -(tr - ca2 / (2.f * mb)) / (2.f * mb);
    const float ssn = sqrtf(ss2);
    // || ss/||ss|| - I/sqrt(C) ||_F^2 = 1 - tr(ss)/(4*||ss||) + 1
    orth[b] = sqrtf(fmaxf(2.f - trss / (4.f * ssn), 0.f));
    csn_s[b] = sqrtf(cs2);
    nvi[b] = 1.0f / nvalid[b];
  }
  __syncthreads();
  if (tid == 0) {
    float sp = 0.f, ot = 0.f, cn = 0.f, ni = 0.f;
    for (int i = 0; i < B_; ++i) { sp += spec[i]; ot += orth[i]; cn += csn_s[i]; ni += nvi[i]; }
    scal_out[0] = sp / (float)B_;                                  // spectral_loss
    scal_out[1] = ot / (float)B_;                                  // ortho_loss
    scal_out[2] = cn * ni * 8.0f / (float)(B_ * B_) - 1.0f;        // cluster_loss
  }
}

extern "C" void kernel_launch(void* const* d_in, const int* in_sizes, int n_in,
                              void* d_out, int out_size, void* d_ws, size_t ws_size,
                              hipStream_t stream) {
  const float* x = (const float*)d_in[0];
  const float* adj = (const float*)d_in[1];
  const unsigned char* mask = (const unsigned char*)d_in[2];  // jnp bool = 1 byte
  const float* W = (const float*)d_in[3];
  const float* bias = (const float*)d_in[4];

  float* out = (float*)d_out;
  float* s_out = out;                                   // [8,2048,64]  = 1,048,576
  float* out_feat = out + (size_t)B_ * N_ * C_;         // [8,64,256]   =   131,072
  float* oadj_out = out_feat + (size_t)B_ * C_ * F_;    // [8,64,64]    =    32,768
  float* scal = oadj_out + (size_t)B_ * C_ * C_;        // 3 scalars

  float* ws = (float*)d_ws;
  float* t = ws;                                        // [8,2048,64]  = 1,048,576
  float* deg = t + (size_t)B_ * N_ * C_;                // [8,2048]     =    16,384
  float* ca = deg + (size_t)B_ * N_;                    // [8,64]
  float* cs = ca + B_ * C_;                             // [8,64]
  float* msum = cs + B_ * C_;                           // [8]
  float* nvalid = msum + B_;                            // [8]
  float* ssm = nvalid + B_;                             // [8,64,64]
  float* oa_raw = ssm + (size_t)B_ * C_ * C_;           // [8,64,64]

  softmax_kernel<<<(B_ * N_) / 32, 256, 0, stream>>>(x, W, bias, mask, s_out);
  adj_s_kernel<<<B_ * 16, 256, 0, stream>>>(adj, s_out, mask, t, deg);
  stx_kernel<<<B_ * 4, 256, 0, stream>>>(s_out, x, out_feat);
  stt_ss_kernel<<<B_ * 2, 256, 0, stream>>>(s_out, t, oa_raw, ssm);
  reduce_kernel<<<B_, 256, 0, stream>>>(s_out, deg, mask, ca, cs, msum, nvalid);
  finalize_kernel<<<1, 256, 0, stream>>>(oa_raw, ssm, ca, cs, msum, nvalid, oadj_out, scal);
}